// AttentionEvaluatorModel_24927990186543
// MI455X (gfx1250) — compile-verified
//
#include <hip/hip_runtime.h>
#include <hip/hip_bf16.h>

// ---------------------------------------------------------------------------
// AttentionEvaluatorModel for MI455X (gfx1250), fp32 WMMA path.
// B=32, Q=16, Fn=4096, E=256, NL=8, L=2.  Memory-bound (~800MB HBM traffic vs
// ~71 GFLOP) -> keep fp32 precision, use V_WMMA_F32_16X16X4_F32 everywhere.
//  - Mappers: 32 rows/WG, B-fragments reused across 2 row-strips (halves L2
//    weight traffic), both residual layers fused in-place in one LDS buffer.
//  - Attention: feature axis split 16-ways per batch (512 WGs) for memory
//    parallelism; partials summed in a small final kernel.
// ---------------------------------------------------------------------------

typedef float v2f __attribute__((ext_vector_type(2)));
typedef float v8f __attribute__((ext_vector_type(8)));

#define E_DIM   256
#define FN      4096
#define B_DIM   32
#define Q_DIM   16
#define NL      8
#define SPLIT   16    // feature-axis splits per batch in the pooling phase
#define LSTR    260   // LDS row stride (floats): 4r mod 64 distinct -> conflict-free
#define WSTR    18    // LDS row stride for 16x16 w-tiles: 18r mod 64 distinct, 8B aligned

// One residual layer y = relu(x @ W + b) + x over R 16-row strips held in one
// LDS buffer, updated IN PLACE (residual folded into registers before the
// barrier, stores after).  Each wave owns two 16-col tiles; a B-fragment is
// loaded once and feeds R WMMAs.
// WMMA fp32 layout (ISA 7.12.2):
//   A (16x4):  lane m<16: {K=k0,k0+1}; lane 16+m: {K=k0+2,k0+3}
//   B (4x16):  vgpr0: lanes0-15 K=k0, lanes16-31 K=k0+2; vgpr1: K=k0+1 / k0+3
//   C (16x16): vgpr r, lane l -> row r+(l>>4)*8, col l&15
template <int R>
__device__ __forceinline__ void residual_layer_wave(
    const float* __restrict__ Wl,   // [256][256] row-major
    const float* __restrict__ bl,   // [256]
    float* xbuf,                    // LDS, R*16 rows, stride LSTR (in/out)
    int lane, int wave)
{
  const int m     = lane & 15;
  const int kh    = (lane >> 4) * 2;
  const int mbase = (lane >> 4) * 8;
  v8f out[2][R];

#pragma unroll
  for (int t = 0; t < 2; ++t) {
    const int n0 = wave * 32 + t * 16;
    const int n  = n0 + m;
    const float bn = bl[n];
    v8f acc[R];
#pragma unroll
    for (int s = 0; s < R; ++s)
#pragma unroll
      for (int r = 0; r < 8; ++r) acc[s][r] = bn;   // C init = bias broadcast

#pragma unroll 4
    for (int k0 = 0; k0 < E_DIM; k0 += 4) {
      v2f bf;                                        // one B frag, R uses
      bf.x = Wl[(k0 + kh)     * E_DIM + n];
      bf.y = Wl[(k0 + kh + 1) * E_DIM + n];
#pragma unroll
      for (int s = 0; s < R; ++s) {
        v2f a;
        a.x = xbuf[(s * 16 + m) * LSTR + k0 + kh];
        a.y = xbuf[(s * 16 + m) * LSTR + k0 + kh + 1];
        acc[s] = __builtin_amdgcn_wmma_f32_16x16x4_f32(
            false, a, false, bf, (short)0, acc[s], false, false);
      }
    }
    // fold relu + residual into registers (reads complete before barrier)
#pragma unroll
    for (int s = 0; s < R; ++s)
#pragma unroll
      for (int r = 0; r < 8; ++r)
        out[t][s][r] = fmaxf(acc[s][r], 0.0f) +
                       xbuf[(s * 16 + mbase + r) * LSTR + n0 + m];
  }

  __syncthreads();   // all waves finished reading xbuf
#pragma unroll
  for (int t = 0; t < 2; ++t)
#pragma unroll
    for (int s = 0; s < R; ++s)
#pragma unroll
      for (int r = 0; r < 8; ++r)
        xbuf[(s * 16 + mbase + r) * LSTR + wave * 32 + t * 16 + m] =
            out[t][s][r];
  __syncthreads();   // stores visible to next layer
}

// Two-layer residual MLP over rows of X. One workgroup (8 waves) per 32-row
// strip pair; both layers fused in a single LDS buffer so HBM traffic is
// read-X + write-Y and each weight element is read once per 32 rows.
__global__ __launch_bounds__(256) void mapper_kernel(
    const float* __restrict__ X, const float* __restrict__ W,
    const float* __restrict__ bias, float* __restrict__ Y)
{
  __shared__ float buf[32 * LSTR];
  const int tid  = threadIdx.x;
  const int wave = tid >> 5, lane = tid & 31;
  const long row0 = (long)blockIdx.x * 32;

#pragma unroll
  for (int i = 0; i < 32; ++i)
    buf[i * LSTR + tid] = X[(row0 + i) * E_DIM + tid];
  __syncthreads();

  for (int layer = 0; layer < 2; ++layer)
    residual_layer_wave<2>(W + layer * E_DIM * E_DIM, bias + layer * E_DIM,
                           buf, lane, wave);

#pragma unroll
  for (int i = 0; i < 32; ++i)
    Y[(row0 + i) * E_DIM + tid] = buf[i * LSTR + tid];
}

// Gated attention + weighted pooling over a 256-feature slice of one batch.
// Grid = B*SPLIT workgroups; each wave owns 2 chunks of 16 feature rows.
// S = q' @ f_chunk^T (WMMA, K=256); gate applied in C-layout; w-tile goes
// through per-wave LDS back into A-layout; pooled accumulated in 16 v8f
// register tiles; ds_add_f32 reduce across waves; partial written to ws.
__global__ __launch_bounds__(256) void attn_pool_kernel(
    const float* __restrict__ qp,    // q' [B,16,256]
    const float* __restrict__ fp,    // f' [B,4096,256]
    const float* __restrict__ vp,    // v' [B,4096,256]
    const float* __restrict__ mask,  // [B,4096]
    const float* __restrict__ ftw,   // [B,4096]
    float* __restrict__ pp)          // pooled partials [B,SPLIT,16,256]
{
  __shared__ float qbuf[16 * LSTR];
  __shared__ float pbuf[16 * LSTR];
  __shared__ float wt[8 * 16 * WSTR];

  const int tid  = threadIdx.x;
  const int wave = tid >> 5, lane = tid & 31;
  const int b    = blockIdx.x / SPLIT;
  const int spl  = blockIdx.x % SPLIT;
  const float* qb = qp + (long)b * Q_DIM * E_DIM;
  const float* fb = fp + (long)b * FN * E_DIM;
  const float* vb = vp + (long)b * FN * E_DIM;

#pragma unroll
  for (int i = 0; i < 16; ++i) qbuf[i * LSTR + tid] = qb[i * E_DIM + tid];
#pragma unroll
  for (int i = 0; i < 16; ++i) pbuf[i * LSTR + tid] = 0.0f;
  __syncthreads();

  v8f pool[16];
#pragma unroll
  for (int t = 0; t < 16; ++t) pool[t] = {};

  const int m  = lane & 15;
  const int kh = (lane >> 4) * 2;
  float* mywt = wt + wave * 16 * WSTR;

  // This WG covers features [spl*256, spl*256+256); 16 chunks, 2 per wave.
  for (int c = 0; c < 2; ++c) {
    const int f0 = spl * (FN / SPLIT) + (c * 8 + wave) * 16;
    // ---- S = q' @ f_chunk^T, K=256 ----
    v8f s = {};
#pragma unroll 4
    for (int k0 = 0; k0 < E_DIM; k0 += 4) {
      v2f a;
      a.x = qbuf[m * LSTR + k0 + kh];
      a.y = qbuf[m * LSTR + k0 + kh + 1];
      const float* frow = fb + (long)(f0 + m) * E_DIM + k0 + kh; // B[k][n]=f'[f0+n][k]
      v2f bf; bf.x = frow[0]; bf.y = frow[1];
      s = __builtin_amdgcn_wmma_f32_16x16x4_f32(
          false, a, false, bf, (short)0, s, false, false);
    }
    // ---- gate: sigmoid(S) * ftw * mask ; spill to LDS w-tile (A-layout src)
    const float g = ftw[b * FN + f0 + m] * mask[b * FN + f0 + m];
    const int mbase = (lane >> 4) * 8;
#pragma unroll
    for (int r = 0; r < 8; ++r) {
      const float sv = s[r];
      mywt[(mbase + r) * WSTR + m] = g / (1.0f + __expf(-sv));
    }
    // ---- pooled += w_tile(16x16) @ v_chunk(16x256), K=16 ----
#pragma unroll
    for (int t = 0; t < 16; ++t) {
      const int e0 = t * 16;
      v8f acc = pool[t];
#pragma unroll
      for (int k0 = 0; k0 < 16; k0 += 4) {
        v2f a;
        a.x = mywt[m * WSTR + k0 + kh];
        a.y = mywt[m * WSTR + k0 + kh + 1];
        const float* vrow = vb + (long)(f0 + k0 + kh) * E_DIM + e0 + m;
        v2f bf; bf.x = vrow[0]; bf.y = vrow[E_DIM];
        acc = __builtin_amdgcn_wmma_f32_16x16x4_f32(
            false, a, false, bf, (short)0, acc, false, false);
      }
      pool[t] = acc;
    }
  }

  // ---- reduce per-wave pooled partials into pbuf via LDS float atomics ----
  {
    const int mbase = (lane >> 4) * 8;
#pragma unroll
    for (int t = 0; t < 16; ++t)
#pragma unroll
      for (int r = 0; r < 8; ++r)
        atomicAdd(&pbuf[(mbase + r) * LSTR + t * 16 + m], pool[t][r]);
  }
  __syncthreads();

  // ---- write this split's partial (coalesced) ----
  float* myp = pp + ((long)b * SPLIT + spl) * Q_DIM * E_DIM;
#pragma unroll
  for (int i = 0; i < 16; ++i)
    myp[i * E_DIM + tid] = pbuf[i * LSTR + tid];
}

// Sum the SPLIT pooled partials, run the Wc residual MLP, project to labels.
// One WG per batch (tiny: ~8MB read total).
__global__ __launch_bounds__(256) void attn_final_kernel(
    const float* __restrict__ pp,    // [B,SPLIT,16,256]
    const float* __restrict__ Wc, const float* __restrict__ bc,
    const float* __restrict__ Wout, const float* __restrict__ bout,
    float* __restrict__ out)         // [B,16,8]
{
  __shared__ float pbuf[16 * LSTR];
  const int tid  = threadIdx.x;
  const int wave = tid >> 5, lane = tid & 31;
  const int b    = blockIdx.x;
  const float* base = pp + (long)b * SPLIT * Q_DIM * E_DIM;

#pragma unroll
  for (int i = 0; i < 16; ++i) {
    float acc = 0.0f;
#pragma unroll
    for (int s = 0; s < SPLIT; ++s)
      acc += base[(s * Q_DIM + i) * E_DIM + tid];
    pbuf[i * LSTR + tid] = acc;
  }
  __syncthreads();

  for (int layer = 0; layer < 2; ++layer)
    residual_layer_wave<1>(Wc + layer * E_DIM * E_DIM, bc + layer * E_DIM,
                           pbuf, lane, wave);

  // ---- out = h @ Wout + bout  (16x256 @ 256x8) ----
  if (tid < Q_DIM * NL) {
    const int mq = tid >> 3, n = tid & 7;
    float acc = bout[n];
#pragma unroll 8
    for (int k = 0; k < E_DIM; ++k)
      acc += pbuf[mq * LSTR + k] * Wout[k * NL + n];
    out[b * Q_DIM * NL + mq * NL + n] = acc;
  }
}

extern "C" void kernel_launch(void* const* d_in, const int* in_sizes, int n_in,
                              void* d_out, int out_size, void* d_ws, size_t ws_size,
                              hipStream_t stream) {
  const float* query = (const float*)d_in[0];
  const float* feats = (const float*)d_in[1];
  const float* vals  = (const float*)d_in[2];
  const float* amask = (const float*)d_in[3];
  const float* ftw   = (const float*)d_in[4];
  const float* Wq    = (const float*)d_in[5];
  const float* bq    = (const float*)d_in[6];
  const float* Wf    = (const float*)d_in[7];
  const float* bf    = (const float*)d_in[8];
  const float* Wv    = (const float*)d_in[9];
  const float* bv    = (const float*)d_in[10];
  const float* Wc    = (const float*)d_in[11];
  const float* bc    = (const float*)d_in[12];
  const float* Wout  = (const float*)d_in[13];
  const float* bout  = (const float*)d_in[14];

  // Workspace: q' (512x256) | f' (131072x256) | v' (131072x256) | partials
  float* qp = (float*)d_ws;
  float* fp = qp + (size_t)B_DIM * Q_DIM * E_DIM;
  float* vp = fp + (size_t)B_DIM * FN * E_DIM;
  float* pp = vp + (size_t)B_DIM * FN * E_DIM;

  mapper_kernel<<<(B_DIM * Q_DIM) / 32, 256, 0, stream>>>(query, Wq, bq, qp);
  mapper_kernel<<<(B_DIM * FN) / 32, 256, 0, stream>>>(feats, Wf, bf, fp);
  mapper_kernel<<<(B_DIM * FN) / 32, 256, 0, stream>>>(vals,  Wv, bv, vp);
  attn_pool_kernel<<<B_DIM * SPLIT, 256, 0, stream>>>(qp, fp, vp, amask, ftw, pp);
  attn_final_kernel<<<B_DIM, 256, 0, stream>>>(pp, Wc, bc, Wout, bout,
                                               (float*)d_out);
}